// GCNReLU_53197464928900
// MI455X (gfx1250) — compile-verified
//
#include <hip/hip_runtime.h>

#define N_NODES  50000
#define N_EDGES  800000
#define N_GRAPHS 128
#define FDIM     128

typedef __attribute__((ext_vector_type(2))) float v2f;
typedef __attribute__((ext_vector_type(8))) float v8f;

// ---------------------------------------------------------------------------
// Dense GEMM  H = X (nrows x 128) @ W (128 x 128)  in fp32 via
// V_WMMA_F32_16X16X4_F32.  Block = 128 threads = 4 waves; block covers a
// 16-row stripe; each wave produces two 16x16 column tiles (reuses A frag).
// A-matrix 16x4 fp32 layout: lanes 0-15 hold K={0,1}, lanes 16-31 K={2,3}
// -> per lane one contiguous float2 load.  C/D: VGPR v -> row v + 8*half.
// ---------------------------------------------------------------------------
__global__ __launch_bounds__(128) void gcn_gemm_f32_wmma(
    const float* __restrict__ X, const float* __restrict__ W,
    float* __restrict__ Hout, int nrows)
{
  const int wave = threadIdx.x >> 5;     // 0..3 -> 32-col super-tile
  const int lane = threadIdx.x & 31;
  const int half = lane >> 4;            // 0 or 1
  const int l    = lane & 15;
  const int rowBase = blockIdx.x << 4;   // 16 rows per block
  if (rowBase >= nrows) return;          // uniform per block
  const int col0 = wave * 32;
  const int col1 = col0 + 16;

  v8f acc0 = {};
  v8f acc1 = {};
  const float* __restrict__ xrow = X + (size_t)(rowBase + l) * FDIM;

#pragma unroll
  for (int k = 0; k < FDIM; k += 4) {
    const int kk = k + 2 * half;                 // this lane's K pair
    const v2f a = *(const v2f*)(xrow + kk);      // global_load_b64
    const float* __restrict__ wr0 = W + (size_t)kk * FDIM;
    const float* __restrict__ wr1 = wr0 + FDIM;
    v2f b0, b1;
    b0.x = wr0[col0 + l];  b0.y = wr1[col0 + l];
    b1.x = wr0[col1 + l];  b1.y = wr1[col1 + l];
    acc0 = __builtin_amdgcn_wmma_f32_16x16x4_f32(false, a, false, b0,
                                                 (short)0, acc0, false, false);
    acc1 = __builtin_amdgcn_wmma_f32_16x16x4_f32(false, a, false, b1,
                                                 (short)0, acc1, false, false);
  }

  float* __restrict__ outb = Hout + (size_t)rowBase * FDIM;
#pragma unroll
  for (int v = 0; v < 8; ++v) {
    const int r = v + 8 * half;
    outb[(size_t)r * FDIM + col0 + l] = acc0[v];
    outb[(size_t)r * FDIM + col1 + l] = acc1[v];
  }
}

// ---------------------------------------------------------------------------
// Degree / normalization:  deg starts at 1 (self loop), count in-edges,
// then dinv = rsqrt(deg) in place.
// ---------------------------------------------------------------------------
__global__ void deg_init(float* __restrict__ deg, int n)
{
  int i = blockIdx.x * blockDim.x + threadIdx.x;
  if (i < n) deg[i] = 1.0f;
}

__global__ void deg_count(const int* __restrict__ dst, float* __restrict__ deg, int e)
{
  int i = blockIdx.x * blockDim.x + threadIdx.x;
  if (i < e) atomicAdd(&deg[dst[i]], 1.0f);
}

__global__ void deg_rsqrt(float* __restrict__ deg, int n)
{
  int i = blockIdx.x * blockDim.x + threadIdx.x;
  if (i < n) deg[i] = rsqrtf(deg[i]);   // deg >= 1 always (self loop)
}

// ---------------------------------------------------------------------------
// Edge scatter:  AGG[dst] += dinv[src]*dinv[dst] * H[src]
// 256 threads handle 2 edges (128 features each); fp32 atomics resolve in L2.
// ---------------------------------------------------------------------------
__global__ __launch_bounds__(256) void gcn_scatter(
    const float* __restrict__ H, const int* __restrict__ src,
    const int* __restrict__ dst, const float* __restrict__ dinv,
    float* __restrict__ AGG, int e)
{
  int edge = blockIdx.x * 2 + (threadIdx.x >> 7);
  int f    = threadIdx.x & (FDIM - 1);
  if (edge >= e) return;
  int s = src[edge];
  int d = dst[edge];
  float nrm = dinv[s] * dinv[d];
  atomicAdd(&AGG[(size_t)d * FDIM + f], H[(size_t)s * FDIM + f] * nrm);
}

// ---------------------------------------------------------------------------
// Self-loop term + bias + optional ReLU, in place on AGG.
// ---------------------------------------------------------------------------
__global__ __launch_bounds__(128) void gcn_self_bias_act(
    const float* __restrict__ H, const float* __restrict__ dinv,
    const float* __restrict__ b, float* __restrict__ AGG, int n, int relu)
{
  int node = blockIdx.x;
  int f    = threadIdx.x;
  if (node >= n) return;
  float di = dinv[node];
  float v = AGG[(size_t)node * FDIM + f]
          + H[(size_t)node * FDIM + f] * di * di
          + b[f];
  if (relu) v = fmaxf(v, 0.0f);
  AGG[(size_t)node * FDIM + f] = v;
}

// ---------------------------------------------------------------------------
// Global mean pool.
// ---------------------------------------------------------------------------
__global__ void pool_count(const int* __restrict__ batch, float* __restrict__ counts, int n)
{
  int i = blockIdx.x * blockDim.x + threadIdx.x;
  if (i < n) atomicAdd(&counts[batch[i]], 1.0f);
}

__global__ __launch_bounds__(128) void pool_sum(
    const float* __restrict__ X, const int* __restrict__ batch,
    float* __restrict__ out, int n)
{
  int node = blockIdx.x;
  int f    = threadIdx.x;
  if (node >= n) return;
  atomicAdd(&out[(size_t)batch[node] * FDIM + f], X[(size_t)node * FDIM + f]);
}

__global__ __launch_bounds__(128) void pool_div(
    float* __restrict__ out, const float* __restrict__ counts)
{
  int g = blockIdx.x;
  int f = threadIdx.x;
  out[(size_t)g * FDIM + f] /= fmaxf(counts[g], 1.0f);
}

// ---------------------------------------------------------------------------
// Orchestration.  Inputs (setup_inputs order):
//  0:x  1:edge_index  2:edge_attr(unused)  3:batch
//  4:W0 5:b0 6:W1 7:b1 8:W2 9:b2 10:W3 11:b3
// ---------------------------------------------------------------------------
extern "C" void kernel_launch(void* const* d_in, const int* in_sizes, int n_in,
                              void* d_out, int out_size, void* d_ws, size_t ws_size,
                              hipStream_t stream)
{
  const float* x     = (const float*)d_in[0];
  const int*   ei    = (const int*)d_in[1];
  const int*   batch = (const int*)d_in[3];
  const float* Wm[4] = {(const float*)d_in[4], (const float*)d_in[6],
                        (const float*)d_in[8], (const float*)d_in[10]};
  const float* bv[4] = {(const float*)d_in[5], (const float*)d_in[7],
                        (const float*)d_in[9], (const float*)d_in[11]};
  const int* src = ei;
  const int* dst = ei + N_EDGES;

  const size_t feat_elems = (size_t)N_NODES * FDIM;
  float* Hbuf   = (float*)d_ws;                 // 25.6 MB
  float* Abuf   = Hbuf + feat_elems;            // 25.6 MB
  float* dinv   = Abuf + feat_elems;            // 200 KB
  float* counts = dinv + N_NODES;               // 512 B

  // --- degree / dinv (graph is fixed across layers) ---
  deg_init <<<(N_NODES + 255) / 256, 256, 0, stream>>>(dinv, N_NODES);
  deg_count<<<(N_EDGES + 255) / 256, 256, 0, stream>>>(dst, dinv, N_EDGES);
  deg_rsqrt<<<(N_NODES + 255) / 256, 256, 0, stream>>>(dinv, N_NODES);

  const float* xin = x;
  for (int layer = 0; layer < 4; ++layer) {
    gcn_gemm_f32_wmma<<<N_NODES / 16, 128, 0, stream>>>(xin, Wm[layer], Hbuf, N_NODES);
    // Abuf may alias xin (layers >= 1); GEMM above has consumed it already.
    hipMemsetAsync(Abuf, 0, feat_elems * sizeof(float), stream);
    gcn_scatter<<<(N_EDGES + 1) / 2, 256, 0, stream>>>(Hbuf, src, dst, dinv, Abuf, N_EDGES);
    gcn_self_bias_act<<<N_NODES, 128, 0, stream>>>(Hbuf, dinv, bv[layer], Abuf,
                                                   N_NODES, layer < 3 ? 1 : 0);
    xin = Abuf;
  }

  // --- global mean pool ---
  hipMemsetAsync(d_out, 0, (size_t)N_GRAPHS * FDIM * sizeof(float), stream);
  hipMemsetAsync(counts, 0, N_GRAPHS * sizeof(float), stream);
  pool_count<<<(N_NODES + 255) / 256, 256, 0, stream>>>(batch, counts, N_NODES);
  pool_sum  <<<N_NODES, 128, 0, stream>>>(Abuf, batch, (float*)d_out, N_NODES);
  pool_div  <<<N_GRAPHS, 128, 0, stream>>>((float*)d_out, counts);
}